// NEUROPULSNxN_2_2_72353019068773
// MI455X (gfx1250) — compile-verified
//
#include <hip/hip_runtime.h>
#include <math.h>

typedef __attribute__((ext_vector_type(2))) float v2f;
typedef __attribute__((ext_vector_type(8))) float v8f;

// Only meaningful on the device pass; host pass has no amdgcn builtins.
#if defined(__HIP_DEVICE_COMPILE__)
#if !__has_builtin(__builtin_amdgcn_wmma_f32_16x16x4_f32)
#error "expected gfx1250 wmma f32 16x16x4 builtin"
#endif
#endif

// ---------------------------------------------------------------------------
// Kernel 1: collapse the 64-stage photonic mesh into one 64x64 complex
// transfer matrix T (x_out = x @ T).  One wave per basis row r of T; lane l
// owns columns (2l, 2l+1).  Substage-1 (wg1 phases on even ports + MMI pairs
// (2l,2l+1)) is lane-local; substage-2 (wg2 phases on odd ports 1..61 + MMI
// pairs (2l+1,2l+2), identity on ports 0 and 63) needs one +/-1 lane shuffle.
// Result stored transposed (n-major) into d_ws: [0..4095]=Re, [4096..8191]=Im.
// ---------------------------------------------------------------------------
__global__ __launch_bounds__(32) void neuropuls_build_T(
    const float* __restrict__ phi1, const float* __restrict__ phi2,
    float* __restrict__ wsT) {
  const int r = blockIdx.x;   // basis row (k-index of T), 0..63
  const int l = threadIdx.x;  // lane, owns cols 2l and 2l+1
  const float S = 0.70710678118654752f;

  float c0r = (2 * l == r) ? 1.f : 0.f, c0i = 0.f;      // col 2l
  float c1r = (2 * l + 1 == r) ? 1.f : 0.f, c1i = 0.f;  // col 2l+1

  for (int s = 0; s < 64; ++s) {
    // ---- substage 1: phase e^{i*phi1[s][2l]} on even port, MMI (2l,2l+1)
    float sn, cs;
    __sincosf(phi1[s * 64 + 2 * l], &sn, &cs);
    float ur = c0r * cs - c0i * sn;
    float ui = c0r * sn + c0i * cs;
    float n0r = S * (ur - c1i), n0i = S * (ui + c1r);   // S*(u + i v)
    float n1r = S * (c1r - ui), n1i = S * (ur + c1i);   // S*(i u + v)
    c0r = n0r; c0i = n0i; c1r = n1r; c1i = n1i;

    // ---- substage 2: phase on odd ports 1..61 (port 63 passthrough),
    //      MMI pairs (2l+1, 2l+2); ports 0 and 63 identity.
    __sincosf(phi2[s * 64 + 2 * l + 1], &sn, &cs);
    float tur = c1r * cs - c1i * sn;
    float tui = c1r * sn + c1i * cs;
    if (l == 31) { tur = c1r; tui = c1i; }  // port 63: no phase

    // shuffles executed by all lanes (wave32)
    float vnr = __shfl_down(c0r, 1);  // lane l gets col 2l+2 (lane l+1's c0)
    float vni = __shfl_down(c0i, 1);
    float upr = __shfl_up(tur, 1);    // lane m gets phased col 2m-1
    float upi = __shfl_up(tui, 1);

    if (l < 31) {                     // new col 2l+1 = S*(u + i v)
      c1r = S * (tur - vni);
      c1i = S * (tui + vnr);
    }
    if (l > 0) {                      // new col 2l = S*(i u_prev + v)
      float t0r = S * (c0r - upi);
      float t0i = S * (c0i + upr);
      c0r = t0r; c0i = t0i;
    }
  }
  // store transposed: wsT[n*64 + k]
  wsT[(2 * l) * 64 + r]            = c0r;
  wsT[4096 + (2 * l) * 64 + r]     = c0i;
  wsT[(2 * l + 1) * 64 + r]        = c1r;
  wsT[4096 + (2 * l + 1) * 64 + r] = c1i;
}

// ---------------------------------------------------------------------------
// Kernel 2: out[b][n] = (x@ReT)^2 + (x@ImT)^2 via V_WMMA_F32_16X16X4_F32.
// 256 threads = 8 waves; T^t (32KB) + 8 x-tiles (32KB) in LDS; each wave
// produces a 16x64 output strip (2 f32 accumulators x 4 N-tiles, K=64 as
// 16 wmma steps each).  Streaming x/out is the bound: ~33.6MB -> ~1.44us.
// ---------------------------------------------------------------------------
__global__ __launch_bounds__(256) void neuropuls_gemm(
    const float* __restrict__ x, const float* __restrict__ wsT,
    float* __restrict__ out) {
  __shared__ float sT[8192];  // [0..4095] Re(T)^t n-major, [4096..8191] Im
  __shared__ float sX[8192];  // 8 waves * (16 rows x 64 cols)
  const int tid = threadIdx.x;
  const int lane = tid & 31;
  const int wv = tid >> 5;

  {  // cooperative load of T^t (broadcast from L2 across all 512 WGs)
    const float4* g = (const float4*)wsT;
    float4* s4 = (float4*)sT;
#pragma unroll
    for (int i = 0; i < 8; ++i) s4[i * 256 + tid] = g[i * 256 + tid];
  }
  {  // per-wave x tile, fully coalesced (tile is a contiguous 4KB span)
    const size_t base = ((size_t)blockIdx.x * 128 + wv * 16) * 64;
    const float4* g = (const float4*)(x + base);
    float4* s4 = (float4*)(sX + wv * 1024);
#pragma unroll
    for (int i = 0; i < 8; ++i) s4[i * 32 + lane] = g[i * 32 + lane];
  }
  __syncthreads();

  const int h = lane >> 4;   // K-half selector (lanes 16..31 hold K+2)
  const int ml = lane & 15;  // M index for A, N index for B/C

  // A fragments for all 16 K-steps: lane holds x[row0+ml][4k+2h .. +1]
  v2f a[16];
  {
    const float2* ax = (const float2*)(sX + wv * 1024 + ml * 64);
#pragma unroll
    for (int k = 0; k < 16; ++k) {
      float2 t = ax[2 * k + h];
      a[k].x = t.x; a[k].y = t.y;
    }
  }

  const size_t row0 = (size_t)blockIdx.x * 128 + wv * 16;
#pragma unroll
  for (int n = 0; n < 4; ++n) {
    const float2* br = (const float2*)(sT + (n * 16 + ml) * 64);
    const float2* bi = (const float2*)(sT + 4096 + (n * 16 + ml) * 64);
    v8f cre = {0.f, 0.f, 0.f, 0.f, 0.f, 0.f, 0.f, 0.f};
    v8f cim = {0.f, 0.f, 0.f, 0.f, 0.f, 0.f, 0.f, 0.f};
#pragma unroll
    for (int k = 0; k < 16; ++k) {
      float2 tr = br[2 * k + h];
      float2 ti = bi[2 * k + h];
      v2f vbr; vbr.x = tr.x; vbr.y = tr.y;
      v2f vbi; vbi.x = ti.x; vbi.y = ti.y;
      cre = __builtin_amdgcn_wmma_f32_16x16x4_f32(false, a[k], false, vbr,
                                                  (short)0, cre, false, false);
      cim = __builtin_amdgcn_wmma_f32_16x16x4_f32(false, a[k], false, vbi,
                                                  (short)0, cim, false, false);
    }
    // detector |.|^2 ; C layout: VGPR v -> row v + 8h, col = n*16 + ml
    float* o = out + (row0 + (size_t)h * 8) * 64 + n * 16 + ml;
#pragma unroll
    for (int v = 0; v < 8; ++v)
      o[(size_t)v * 64] = cre[v] * cre[v] + cim[v] * cim[v];
  }
}

extern "C" void kernel_launch(void* const* d_in, const int* in_sizes, int n_in,
                              void* d_out, int out_size, void* d_ws,
                              size_t ws_size, hipStream_t stream) {
  const float* x = (const float*)d_in[0];      // (65536, 64)
  const float* phi1 = (const float*)d_in[1];   // (64, 64)
  const float* phi2 = (const float*)d_in[2];   // (64, 64)
  float* wsT = (float*)d_ws;                   // 8192 floats = 32 KB
  float* out = (float*)d_out;                  // (65536, 64) f32

  neuropuls_build_T<<<dim3(64), dim3(32), 0, stream>>>(phi1, phi2, wsT);
  neuropuls_gemm<<<dim3(512), dim3(256), 0, stream>>>(x, wsT, out);
}